// SimpleBetterGCN_52201032515746
// MI455X (gfx1250) — compile-verified
//
#include <hip/hip_runtime.h>
#include <hip/hip_bf16.h>
#include <math.h>

typedef __attribute__((ext_vector_type(2))) float v2f;
typedef __attribute__((ext_vector_type(8))) float v8f;

#define HID 128

// ---------------------------------------------------------------------------
// Small init kernels
// ---------------------------------------------------------------------------
__global__ void fill_kernel(float* __restrict__ p, float v, size_t n) {
  size_t i = (size_t)blockIdx.x * blockDim.x + threadIdx.x;
  size_t stride = (size_t)gridDim.x * blockDim.x;
  for (; i < n; i += stride) p[i] = v;
}

__global__ void init_stats_kernel(float* __restrict__ stats) {
  if (threadIdx.x == 0) {
    stats[0] = -INFINITY;  // running max
    stats[1] = 0.0f;       // running sum of exp
  }
}

// ---------------------------------------------------------------------------
// CSR row_ptr from sorted COO rows: row_ptr[r] = lower_bound(rows, r).
// row_ptr[N] = E falls out naturally. One thread per row (incl. r == N).
// ---------------------------------------------------------------------------
__global__ void rowptr_kernel(const int* __restrict__ rows,
                              int* __restrict__ row_ptr, int n, int nedges) {
  const int r = blockIdx.x * blockDim.x + threadIdx.x;
  if (r > n) return;
  int lo = 0, hi = nedges;
  while (lo < hi) {
    int mid = (lo + hi) >> 1;
    if (rows[mid] < r) lo = mid + 1; else hi = mid;
  }
  row_ptr[r] = lo;
}

// ---------------------------------------------------------------------------
// GEMM + bias: out[N x 128] = A[N x K] @ W[K x 128] + bias[128]
// One block = 16 rows x 128 cols. 8 waves; wave w owns 16x16 tile at col 16w.
// f32 WMMA 16x16x4, A tile staged in LDS.
// ---------------------------------------------------------------------------
template <int K>
__global__ __launch_bounds__(256) void gemm_bias_kernel(
    const float* __restrict__ A, const float* __restrict__ W,
    const float* __restrict__ bias, float* __restrict__ out, int nrows) {
  __shared__ float As[16 * (K + 1)];  // +1 pad: rows on distinct LDS banks

  const int tid = threadIdx.x;
  const int r0  = blockIdx.x * 16;

  for (int i = tid; i < 16 * K; i += 256) {
    int r = i / K, k = i - r * K;
    int row = r0 + r;
    As[r * (K + 1) + k] = (row < nrows) ? A[(size_t)row * K + k] : 0.0f;
  }
  __syncthreads();

  const int wave = tid >> 5;   // 0..7
  const int lane = tid & 31;
  const int half = lane >> 4;  // 0 or 1
  const int lr   = lane & 15;
  const int c0   = wave * 16;

  v8f acc = 0.0f;
#pragma unroll
  for (int k0 = 0; k0 < K; k0 += 4) {
    const int ka = k0 + 2 * half;
    v2f a, b;
    // A 16x4: lane = M; VGPR v holds K = 2*half + v
    a[0] = As[lr * (K + 1) + ka];
    a[1] = As[lr * (K + 1) + ka + 1];
    // B 4x16: lane = N; VGPR v holds K = 2*half + v
    b[0] = W[(size_t)ka * HID + c0 + lr];
    b[1] = W[(size_t)(ka + 1) * HID + c0 + lr];
    acc = __builtin_amdgcn_wmma_f32_16x16x4_f32(
        false, a, false, b, (short)0, acc, false, false);
  }

  const float bv = bias[c0 + lr];
#pragma unroll
  for (int i = 0; i < 8; ++i) {
    int row = r0 + i + 8 * half;  // C/D: M = vgpr + 8*(lane>=16), N = lane%16
    if (row < nrows) out[(size_t)row * HID + c0 + lr] = acc[i] + bv;
  }
}

// ---------------------------------------------------------------------------
// CSR SpMM, wave per row, no atomics, fused epilogue.
//   MODE 0: out[row] = acc
//   MODE 1: out[row] = relu(acc)                       (produces h1)
//   MODE 2: out[row] = extra[row] + relu(acc)          (produces h = h1 + h2)
// Each of the 32 lanes owns 4 consecutive columns -> global_load_b128 gathers.
// ---------------------------------------------------------------------------
template <int MODE>
__global__ __launch_bounds__(256) void spmm_csr_kernel(
    const int* __restrict__ row_ptr, const int* __restrict__ cols,
    const float* __restrict__ vals, const float* __restrict__ in,
    const float* __restrict__ extra, float* __restrict__ out, int n) {
  const int wave = threadIdx.x >> 5;
  const int lane = threadIdx.x & 31;
  const int row = blockIdx.x * 8 + wave;
  if (row >= n) return;

  const int start = row_ptr[row];
  const int end   = row_ptr[row + 1];

  const float4* __restrict__ in4 = (const float4*)in;
  float4 acc = make_float4(0.0f, 0.0f, 0.0f, 0.0f);
  for (int e = start; e < end; ++e) {
    const int   c = cols[e];   // wave-uniform
    const float v = vals[e];
    const float4 xv = in4[(size_t)c * (HID / 4) + lane];
    acc.x += v * xv.x;
    acc.y += v * xv.y;
    acc.z += v * xv.z;
    acc.w += v * xv.w;
  }

  if (MODE >= 1) {
    acc.x = fmaxf(acc.x, 0.0f);
    acc.y = fmaxf(acc.y, 0.0f);
    acc.z = fmaxf(acc.z, 0.0f);
    acc.w = fmaxf(acc.w, 0.0f);
  }
  if (MODE == 2) {
    const float4 hv = ((const float4*)extra)[(size_t)row * (HID / 4) + lane];
    acc.x += hv.x; acc.y += hv.y; acc.z += hv.z; acc.w += hv.w;
  }
  ((float4*)out)[(size_t)row * (HID / 4) + lane] = acc;
}

// ---------------------------------------------------------------------------
// Attention scores: scores[row] = dot(h[row], att_w) + att_b. Wave per row.
// ---------------------------------------------------------------------------
__global__ __launch_bounds__(256) void att_kernel(
    const float* __restrict__ h, const float* __restrict__ att_w,
    const float* __restrict__ att_b, float* __restrict__ scores, int n) {
  const int wave = threadIdx.x >> 5;
  const int lane = threadIdx.x & 31;
  const int row = blockIdx.x * 8 + wave;
  if (row >= n) return;
  float s = 0.0f;
#pragma unroll
  for (int c = lane; c < HID; c += 32) s += h[(size_t)row * HID + c] * att_w[c];
#pragma unroll
  for (int off = 16; off > 0; off >>= 1) s += __shfl_xor(s, off, 32);
  if (lane == 0) scores[row] = s + att_b[0];
}

// ---------------------------------------------------------------------------
// Reductions for softmax over N
// ---------------------------------------------------------------------------
__device__ inline void atomicMaxF(float* addr, float val) {
  unsigned int* ua = (unsigned int*)addr;
  unsigned int old = *ua;
  while (__uint_as_float(old) < val) {
    unsigned int assumed = old;
    old = atomicCAS(ua, assumed, __float_as_uint(val));
    if (old == assumed) break;
  }
}

__global__ __launch_bounds__(256) void max_kernel(const float* __restrict__ s,
                                                  float* __restrict__ stats,
                                                  int n) {
  __shared__ float red[8];
  float m = -INFINITY;
  for (size_t i = (size_t)blockIdx.x * blockDim.x + threadIdx.x; i < (size_t)n;
       i += (size_t)gridDim.x * blockDim.x)
    m = fmaxf(m, s[i]);
#pragma unroll
  for (int off = 16; off > 0; off >>= 1) m = fmaxf(m, __shfl_xor(m, off, 32));
  const int wave = threadIdx.x >> 5, lane = threadIdx.x & 31;
  if (lane == 0) red[wave] = m;
  __syncthreads();
  if (threadIdx.x == 0) {
    float bm = red[0];
    for (int w = 1; w < 8; ++w) bm = fmaxf(bm, red[w]);
    atomicMaxF(&stats[0], bm);
  }
}

__global__ __launch_bounds__(256) void sumexp_kernel(
    const float* __restrict__ s, float* __restrict__ stats, int n) {
  __shared__ float red[8];
  const float m = stats[0];
  float acc = 0.0f;
  for (size_t i = (size_t)blockIdx.x * blockDim.x + threadIdx.x; i < (size_t)n;
       i += (size_t)gridDim.x * blockDim.x)
    acc += __expf(s[i] - m);
#pragma unroll
  for (int off = 16; off > 0; off >>= 1) acc += __shfl_xor(acc, off, 32);
  const int wave = threadIdx.x >> 5, lane = threadIdx.x & 31;
  if (lane == 0) red[wave] = acc;
  __syncthreads();
  if (threadIdx.x == 0) {
    float bs = 0.0f;
    for (int w = 0; w < 8; ++w) bs += red[w];
    atomicAdd(&stats[1], bs);
  }
}

// g[c] += sum_rows exp(s[row]-m) * h[row][c]  (unnormalized; /Z at the end)
__global__ __launch_bounds__(256) void accum_g_kernel(
    const float* __restrict__ h, const float* __restrict__ scores,
    const float* __restrict__ stats, float* __restrict__ g, int n) {
  __shared__ float part[HID];
  for (int i = threadIdx.x; i < HID; i += 256) part[i] = 0.0f;
  __syncthreads();
  const float m = stats[0];
  const int wave = threadIdx.x >> 5, lane = threadIdx.x & 31;
  for (int row = blockIdx.x * 8 + wave; row < n; row += gridDim.x * 8) {
    const float w = __expf(scores[row] - m);
#pragma unroll
    for (int c = lane; c < HID; c += 32)
      atomicAdd(&part[c], w * h[(size_t)row * HID + c]);  // LDS ds_add_f32
  }
  __syncthreads();
  for (int i = threadIdx.x; i < HID; i += 256) atomicAdd(&g[i], part[i]);
}

// out[c] = cls_b[c] + sum_j (g[j]/Z) * cls_w[j][c]
__global__ void final_kernel(const float* __restrict__ g,
                             const float* __restrict__ stats,
                             const float* __restrict__ cls_w,
                             const float* __restrict__ cls_b,
                             float* __restrict__ out, int nclasses) {
  const int c = threadIdx.x;
  if (c < nclasses) {
    const float invZ = 1.0f / stats[1];
    float acc = cls_b[c];
    for (int j = 0; j < HID; ++j)
      acc += (g[j] * invZ) * cls_w[j * nclasses + c];
    out[c] = acc;
  }
}

// ---------------------------------------------------------------------------
// Launch
// ---------------------------------------------------------------------------
extern "C" void kernel_launch(void* const* d_in, const int* in_sizes, int n_in,
                              void* d_out, int out_size, void* d_ws,
                              size_t ws_size, hipStream_t stream) {
  const float* x      = (const float*)d_in[0];
  const int*   erows  = (const int*)d_in[1];
  const int*   ecols  = (const int*)d_in[2];
  const float* evals  = (const float*)d_in[3];
  const float* fc1_w  = (const float*)d_in[4];
  const float* fc1_b  = (const float*)d_in[5];
  const float* fc2_w  = (const float*)d_in[6];
  const float* fc2_b  = (const float*)d_in[7];
  const float* att_w  = (const float*)d_in[8];
  const float* att_b  = (const float*)d_in[9];
  const float* cls_w  = (const float*)d_in[10];
  const float* cls_b  = (const float*)d_in[11];
  float* out = (float*)d_out;

  const int N = in_sizes[0] / 64;   // 100000
  const int E = in_sizes[1];        // 3200000
  const size_t NH = (size_t)N * HID;

  // Workspace layout (floats, all 16B aligned)
  float* B0      = (float*)d_ws;    // gemm output (P1, then P2)
  float* B1      = B0 + NH;         // h1 = relu(spmm(P1))
  float* B2      = B1 + NH;         // h  = h1 + relu(spmm(P2))
  float* scores  = B2 + NH;         // [N]
  float* g       = scores + N;      // [128]
  float* stats   = g + HID;         // [2]: max, Z
  int*   row_ptr = (int*)(stats + 2);  // [N+1]

  const int row_tiles = (N + 15) / 16;  // 6250
  const int row_waves = (N + 7) / 8;    // wave-per-row blocks

  // Init + CSR build (only tiny buffers need zeroing now)
  fill_kernel<<<1, 128, 0, stream>>>(g, 0.0f, (size_t)HID);
  init_stats_kernel<<<1, 32, 0, stream>>>(stats);
  rowptr_kernel<<<(N + 256) / 256, 256, 0, stream>>>(erows, row_ptr, N, E);

  // Layer 1: P1 = X@W1 + b1 ; h1 = relu(A @ P1)  (relu fused into spmm store)
  gemm_bias_kernel<64><<<row_tiles, 256, 0, stream>>>(x, fc1_w, fc1_b, B0, N);
  spmm_csr_kernel<1><<<row_waves, 256, 0, stream>>>(row_ptr, ecols, evals, B0,
                                                    nullptr, B1, N);

  // Layer 2: P2 = h1@W2 + b2 ; h = h1 + relu(A @ P2) (fused into spmm store)
  gemm_bias_kernel<128><<<row_tiles, 256, 0, stream>>>(B1, fc2_w, fc2_b, B0, N);
  spmm_csr_kernel<2><<<row_waves, 256, 0, stream>>>(row_ptr, ecols, evals, B0,
                                                    B1, B2, N);

  // Attention pooling: softmax over nodes, weighted feature sum
  att_kernel<<<row_waves, 256, 0, stream>>>(B2, att_w, att_b, scores, N);
  max_kernel<<<1024, 256, 0, stream>>>(scores, stats, N);
  sumexp_kernel<<<1024, 256, 0, stream>>>(scores, stats, N);
  accum_g_kernel<<<512, 256, 0, stream>>>(B2, scores, stats, g, N);

  // Classifier
  final_kernel<<<1, 32, 0, stream>>>(g, stats, cls_w, cls_b, out, 10);
}